// gru_module_1_10264971837855
// MI455X (gfx1250) — compile-verified
//
#include <hip/hip_runtime.h>

typedef __attribute__((ext_vector_type(2))) float v2f;
typedef __attribute__((ext_vector_type(4))) float v4f;
typedef __attribute__((ext_vector_type(8))) float v8f;

#define T24   24
#define NSEG  276          // T*(T-1)/2
#define MPAD  288          // 18 * 16 (padded M so stores are unconditional)
#define KDIM  72
#define KTILES 18          // 72 / 4
#define ETA_C 0.95f
#define ETA_D 0.95f

// Single-workgroup kernel: 288 threads = 9 wave32's.
// Wave w computes M-tiles {2w, 2w+1} of X1 = W @ X with v_wmma_f32_16x16x4_f32.
__global__ __launch_bounds__(288)
void ragged_softmin_kernel(const float* __restrict__ pi_s,
                           const float* __restrict__ pi_c,
                           const float* __restrict__ pi_g,
                           const float* __restrict__ Bp,
                           const float* __restrict__ W,
                           float* __restrict__ out) {
  __shared__ __align__(16) float Xs[KDIM];   // X = [pi_s/eta_d, pi_c/(eta_c*eta_d), pi_g]
  __shared__ __align__(16) float X1s[MPAD];  // flat ragged segments (276 live + 12 pad)

  const int tid = threadIdx.x;

  // ---- phase 0: build X in LDS ----
  if (tid < T24) {
    Xs[tid]        = pi_s[tid] * (1.0f / ETA_D);
    Xs[tid + 24]   = pi_c[tid] * (1.0f / (ETA_C * ETA_D));
    Xs[tid + 48]   = pi_g[tid];
  }
  __syncthreads();

  // ---- phase 1: X1 = W @ X via WMMA f32 16x16x4 ----
  // A (16x4) lane layout per ISA: lanes 0-15 hold K=0,1 in v0,v1 (row M=lane);
  // lanes 16-31 hold K=2,3 (row M=lane-16).
  // B (4x16): X[k0+k] replicated across all 16 columns -> every D column is a
  // copy; lane supplies v0 = K0(lo half)/K2(hi), v1 = K1(lo)/K3(hi).
  //
  // Out-of-range A rows (276..287, last M-tile only) corrupt ONLY their own
  // D rows, which land in X1s' pad slots — so we CLAMP the load address and
  // store unconditionally. No exec save/restore anywhere in the hot path.
  const int wave = tid >> 5;
  const int lane = tid & 31;
  const int lhi  = lane >> 4;   // half-wave: 0 or 1
  const int lm   = lane & 15;   // row-within-tile

  for (int s = 0; s < 2; ++s) {
    const int mt = wave * 2 + s;          // 0..17
    const int m0 = mt * 16;
    const int row = m0 + lm;
    const int rowc = (row < NSEG) ? row : (NSEG - 1);   // clamp, hoisted
    const float* __restrict__ wrow = W + rowc * KDIM + (lhi << 1);
    const float* __restrict__ xcol = Xs + (lhi << 1);

    v8f acc = {};
    #pragma unroll
    for (int kt = 0; kt < KTILES; ++kt) {
      const v2f a = *(const v2f*)(wrow + kt * 4);   // global_load_b64
      const v2f b = *(const v2f*)(xcol + kt * 4);   // ds_load_b64 / 2addr
      acc = __builtin_amdgcn_wmma_f32_16x16x4_f32(
          /*neg_a=*/false, a, /*neg_b=*/false, b,
          /*c_mod=*/(short)0, acc, /*reuse_a=*/false, /*reuse_b=*/false);
    }
    // Every lane holds a full copy of its 16-row result column (B replicated):
    // vgpr r, lanes 0-15 -> M = m0+r ; lanes 16-31 -> M = m0+8+r.
    // Unconditional 2x b128 store into the padded X1s.
    if (lm == 0) {
      float* dst = X1s + m0 + (lhi ? 8 : 0);        // 32-byte aligned
      v4f lo = { acc[0], acc[1], acc[2], acc[3] };
      v4f hi = { acc[4], acc[5], acc[6], acc[7] };
      *(v4f*)(dst)     = lo;   // ds_store_b128
      *(v4f*)(dst + 4) = hi;   // ds_store_b128
    }
  }
  __syncthreads();

  // ---- phase 2: per-row min (X2) and stable softmin (X3) ----
  if (tid < T24) {
    const int t = tid;
    const float Bv = Bp[0];
    const int base = (t * (t - 1)) >> 1;

    float mn = __builtin_inff();
    for (int i = 0; i < t; ++i) mn = fminf(mn, X1s[base + i]);
    out[t] = mn;  // X2 (row 0 -> +inf, matches jnp.min over empty/inf fill)

    float sum = 0.0f;
    for (int i = 0; i < t; ++i)
      sum += __expf(-Bv * (X1s[base + i] - mn));
    const float inv = (t > 0) ? (1.0f / sum) : 0.0f;

    float* rowp = out + T24 + t * T24;
    for (int i = 0; i < T24; ++i) {
      float v = 0.0f;
      if (i < t) v = __expf(-Bv * (X1s[base + i] - mn)) * inv;
      rowp[i] = v;
    }
  }
}

extern "C" void kernel_launch(void* const* d_in, const int* in_sizes, int n_in,
                              void* d_out, int out_size, void* d_ws, size_t ws_size,
                              hipStream_t stream) {
  const float* pi_s = (const float*)d_in[0];
  const float* pi_c = (const float*)d_in[1];
  const float* pi_g = (const float*)d_in[2];
  const float* B    = (const float*)d_in[3];
  const float* W    = (const float*)d_in[4];
  float* out = (float*)d_out;

  hipLaunchKernelGGL(ragged_softmin_kernel, dim3(1), dim3(288), 0, stream,
                     pi_s, pi_c, pi_g, B, W, out);
}